// CAFBlock_9663676416189
// MI455X (gfx1250) — compile-verified
//
#include <hip/hip_runtime.h>

// ---------------- problem dims ----------------
#define CA    128
#define TT    512
#define FQd   128
#define TVd   64
#define NSAMP 4                 // B*NS
#define TFd   (TT*FQd)          // 65536 elems per (b,c) plane
#define CTFd  (CA*TFd)          // 8388608 elems per batch of a1
// workspace layout (float units)
#define WS_PART  0              // 128*32 float2  -> 8192 floats
#define WS_COEF  8192           // 4*128 floats
#define WS_VKEY  8704           // 4*128*64
#define WS_VATT  41472          // 4*128*64

typedef float v4f __attribute__((ext_vector_type(4)));

// low 32 bits of a flat shared-memory address == LDS byte offset (ISA 10.2)
__device__ __forceinline__ unsigned lds_off(const void* p) {
  return (unsigned)(unsigned long long)p;
}

// CDNA5 async global->LDS copy, 16B per lane (one b128 row stage per wave)
__device__ __forceinline__ void async_row_ld16(unsigned lds_byte_addr, const float* gptr) {
  asm volatile("global_load_async_to_lds_b128 %0, %1, off"
               :: "v"(lds_byte_addr), "v"(gptr) : "memory");
}

// ---------------- K1: per-channel partial sums of a1 ----------------
__global__ void bn_partials_k(const float* __restrict__ a1, float2* __restrict__ part) {
  const int c = blockIdx.y;           // 0..127
  const int q = blockIdx.x;           // 0..31 chunks per channel
  const int b = q >> 4, p = q & 15;   // 16 chunks per (b,c) plane
  const int tid = threadIdx.x;
  const float4* base =
      (const float4*)(a1 + (size_t)b * CTFd + (size_t)c * TFd + (size_t)p * 4096);
  float s = 0.f, ss = 0.f;
#pragma unroll
  for (int k = 0; k < 4; ++k) {
    float4 x = base[tid + k * 256];
    s  += x.x + x.y + x.z + x.w;
    ss += x.x * x.x + x.y * x.y + x.z * x.z + x.w * x.w;
  }
  __shared__ float r0[256], r1[256];
  r0[tid] = s; r1[tid] = ss;
  __syncthreads();
  for (int off = 128; off > 0; off >>= 1) {
    if (tid < off) { r0[tid] += r0[tid + off]; r1[tid] += r1[tid + off]; }
    __syncthreads();
  }
  if (tid == 0) part[c * 32 + q] = make_float2(r0[0], r1[0]);
}

// ---------------- K2: fold BN stats into per-channel affine coeffs ----------------
__global__ void bn_finalize_k(const float2* __restrict__ part,
                              const float* __restrict__ w_val,  const float* __restrict__ b_val,
                              const float* __restrict__ g_val,  const float* __restrict__ be_val,
                              const float* __restrict__ w_gate, const float* __restrict__ b_gate,
                              const float* __restrict__ g_gate, const float* __restrict__ be_gate,
                              float* __restrict__ coef) {
  const int c = threadIdx.x;          // 128 threads
  float S = 0.f, Q = 0.f;
  for (int q = 0; q < 32; ++q) { float2 p = part[c * 32 + q]; S += p.x; Q += p.y; }
  const float invN = 1.0f / 131072.0f;
  const float mu  = S * invN;
  const float var = Q * invN - mu * mu;
  // bn2d(a1*w + b) stats: mean' = w*mu + b, var' = w^2*var  ->  affine in a1
  {
    const float rs = rsqrtf(w_val[c] * w_val[c] * var + 1e-5f);
    coef[c]       = w_val[c] * g_val[c] * rs;                       // A_val
    coef[128 + c] = be_val[c] - w_val[c] * mu * g_val[c] * rs;      // B_val
  }
  {
    const float rs = rsqrtf(w_gate[c] * w_gate[c] * var + 1e-5f);
    coef[256 + c] = w_gate[c] * g_gate[c] * rs;                     // A_gate
    coef[384 + c] = be_gate[c] - w_gate[c] * mu * g_gate[c] * rs;   // B_gate
  }
}

// ---------------- K3: video paths (gln key + softmax attn), Tv resolution ----------------
__global__ void v_paths_k(const float* __restrict__ v1,
                          const float* __restrict__ w_key,  const float* __restrict__ b_key,
                          const float* __restrict__ g_key,  const float* __restrict__ be_key,
                          const float* __restrict__ w_attn, const float* __restrict__ b_attn,
                          const float* __restrict__ g_attn, const float* __restrict__ be_attn,
                          float* __restrict__ vkey_ws, float* __restrict__ vattn_ws) {
  const int n   = blockIdx.x;          // sample 0..3
  const int tid = threadIdx.x;
  const float* v = v1 + n * (CA * TVd);

  // --- stats for key-pre (C*Tv=8192) and attn-pre (C*H*Tv=32768) ---
  float kS = 0.f, kQ = 0.f, aS = 0.f, aQ = 0.f;
  for (int j = tid; j < CA * TVd; j += 256) {
    const int c = j >> 6;
    const float x = v[j];
    const float y = x * w_key[c] + b_key[c];
    kS += y; kQ += y * y;
#pragma unroll
    for (int h = 0; h < 4; ++h) {
      const int jj = c * 4 + h;
      const float z = x * w_attn[jj] + b_attn[jj];
      aS += z; aQ += z * z;
    }
  }
  __shared__ float s0[256], s1[256], s2[256], s3[256];
  s0[tid] = kS; s1[tid] = kQ; s2[tid] = aS; s3[tid] = aQ;
  __syncthreads();
  for (int off = 128; off > 0; off >>= 1) {
    if (tid < off) {
      s0[tid] += s0[tid + off]; s1[tid] += s1[tid + off];
      s2[tid] += s2[tid + off]; s3[tid] += s3[tid + off];
    }
    __syncthreads();
  }
  __shared__ float sh_mkey, sh_rskey, sh_rsattn;
  if (tid == 0) {
    const float mk = s0[0] / 8192.0f;
    const float vk = s1[0] / 8192.0f - mk * mk;
    const float ma = s2[0] / 32768.0f;
    const float va = s3[0] / 32768.0f - ma * ma;
    sh_mkey   = mk;
    sh_rskey  = rsqrtf(vk + 1e-8f);
    sh_rsattn = rsqrtf(va + 1e-8f);
  }
  __syncthreads();
  const float mkey = sh_mkey, rskey = sh_rskey, rsattn = sh_rsattn;

  // --- v_key at Tv resolution ---
  for (int j = tid; j < CA * TVd; j += 256) {
    const int c = j >> 6;
    const float x = v[j];
    vkey_ws[n * (CA * TVd) + j] =
        (x * w_key[c] + b_key[c] - mkey) * rskey * g_key[c] + be_key[c];
  }

  // --- attn: vm is affine in v1; constant offset cancels in softmax over tv ---
  if (tid < CA) {
    const int c = tid;
    const float K = 0.25f * (w_attn[4 * c + 0] * g_attn[4 * c + 0] +
                             w_attn[4 * c + 1] * g_attn[4 * c + 1] +
                             w_attn[4 * c + 2] * g_attn[4 * c + 2] +
                             w_attn[4 * c + 3] * g_attn[4 * c + 3]);
    const float Sc = rsattn * K;
    const float* row = v + c * TVd;
    float mx = -3.4e38f;
    for (int t = 0; t < TVd; ++t) mx = fmaxf(mx, Sc * row[t]);
    float sum = 0.f;
    for (int t = 0; t < TVd; ++t) sum += expf(Sc * row[t] - mx);
    const float inv = 1.0f / sum;
    float* orow = vattn_ws + n * (CA * TVd) + c * TVd;
    for (int t = 0; t < TVd; ++t) orow[t] = expf(Sc * row[t] - mx) * inv;
  }
}

// ---------------- K4: bandwidth-bound fuse with async global->LDS pipeline ----------------
// rows = B*C*T = 131072; each row = 128 contiguous floats of a1 -> 2 output rows.
// A wave's 16 rows share (b,c) and span exactly two tv buckets -> hoist all scalars.
#define ROWS_PER_WAVE 16
__global__ void fuse_k(const float* __restrict__ a1,
                       const float* __restrict__ coef,
                       const float* __restrict__ vkey,
                       const float* __restrict__ vattn,
                       float* __restrict__ out) {
  __shared__ __align__(16) float lbuf[8 * 2 * 128];   // 8 waves x 2 buffers x 512B
  const int wv   = threadIdx.x >> 5;
  const int lane = threadIdx.x & 31;
  const unsigned row0 = (blockIdx.x * 8u + wv) * ROWS_PER_WAVE;

  // wave-invariant decode: row = ((b*128)+c)*512 + t ; t spans [t0, t0+16)
  const unsigned b  = row0 >> 16;
  const unsigned c  = (row0 >> 9) & 127u;
  const unsigned t0 = row0 & 511u;
  const float Av = coef[c],       Bv = coef[128 + c];
  const float Ag = coef[256 + c], Bg = coef[384 + c];
  const unsigned n0 = b * 2u, n1 = n0 + 1u;
  const unsigned vb0 = n0 * (CA * TVd) + c * TVd;
  const unsigned vb1 = n1 * (CA * TVd) + c * TVd;
  const unsigned tv0 = t0 >> 3;                 // rows 0..7 -> tv0, rows 8..15 -> tv0+1
  const float at0a = vattn[vb0 + tv0],     at1a = vattn[vb1 + tv0];
  const float ky0a = vkey [vb0 + tv0],     ky1a = vkey [vb1 + tv0];
  const float at0b = vattn[vb0 + tv0 + 1], at1b = vattn[vb1 + tv0 + 1];
  const float ky0b = vkey [vb0 + tv0 + 1], ky1b = vkey [vb1 + tv0 + 1];

  float* myb = &lbuf[wv * 256];
  const unsigned lds0 = lds_off(myb) + (unsigned)lane * 16u;   // buffer 0, this lane
  const float* gbase  = a1 + (size_t)row0 * FQd + lane * 4;
  const size_t ob0    = ((size_t)n0 * CA + c) * TFd + (size_t)t0 * FQd + lane * 4;
  const size_t ob1    = ((size_t)n1 * CA + c) * TFd + (size_t)t0 * FQd + lane * 4;

  // prologue: stage row 0 into buffer 0
  async_row_ld16(lds0, gbase);

#pragma unroll
  for (int i = 0; i < ROWS_PER_WAVE; ++i) {
    if (i + 1 < ROWS_PER_WAVE) {
      async_row_ld16(lds0 + (unsigned)(((i + 1) & 1) * 512), gbase + (i + 1) * FQd);
      asm volatile("s_wait_asynccnt 1" ::: "memory");   // row i landed
    } else {
      asm volatile("s_wait_asynccnt 0" ::: "memory");
    }

    const float at0 = (i < 8) ? at0a : at0b;
    const float at1 = (i < 8) ? at1a : at1b;
    const float ky0 = (i < 8) ? ky0a : ky0b;
    const float ky1 = (i < 8) ? ky1a : ky1b;

    const v4f x = *(const v4f*)&myb[(i & 1) * 128 + lane * 4];

    v4f o0, o1;
    {
      float av, ag;
      av = Av * x.x + Bv; ag = fmaxf(Ag * x.x + Bg, 0.f);
      o0.x = at0 * av + ky0 * ag; o1.x = at1 * av + ky1 * ag;
      av = Av * x.y + Bv; ag = fmaxf(Ag * x.y + Bg, 0.f);
      o0.y = at0 * av + ky0 * ag; o1.y = at1 * av + ky1 * ag;
      av = Av * x.z + Bv; ag = fmaxf(Ag * x.z + Bg, 0.f);
      o0.z = at0 * av + ky0 * ag; o1.z = at1 * av + ky1 * ag;
      av = Av * x.w + Bv; ag = fmaxf(Ag * x.w + Bg, 0.f);
      o0.w = at0 * av + ky0 * ag; o1.w = at1 * av + ky1 * ag;
    }
    // non-temporal: 128MB write-once stream must not evict L2-resident a1
    __builtin_nontemporal_store(o0, (v4f*)(out + ob0 + (size_t)i * FQd));
    __builtin_nontemporal_store(o1, (v4f*)(out + ob1 + (size_t)i * FQd));
  }
}

// ---------------- launch ----------------
extern "C" void kernel_launch(void* const* d_in, const int* in_sizes, int n_in,
                              void* d_out, int out_size, void* d_ws, size_t ws_size,
                              hipStream_t stream) {
  const float* a1      = (const float*)d_in[0];
  const float* v1      = (const float*)d_in[1];
  const float* w_gate  = (const float*)d_in[2];
  const float* b_gate  = (const float*)d_in[3];
  const float* g_gate  = (const float*)d_in[4];
  const float* be_gate = (const float*)d_in[5];
  const float* w_val   = (const float*)d_in[6];
  const float* b_val   = (const float*)d_in[7];
  const float* g_val   = (const float*)d_in[8];
  const float* be_val  = (const float*)d_in[9];
  const float* w_attn  = (const float*)d_in[10];
  const float* b_attn  = (const float*)d_in[11];
  const float* g_attn  = (const float*)d_in[12];
  const float* be_attn = (const float*)d_in[13];
  const float* w_key   = (const float*)d_in[14];
  const float* b_key   = (const float*)d_in[15];
  const float* g_key   = (const float*)d_in[16];
  const float* be_key  = (const float*)d_in[17];

  float*  ws    = (float*)d_ws;
  float2* part  = (float2*)(ws + WS_PART);
  float*  coef  = ws + WS_COEF;
  float*  vkey  = ws + WS_VKEY;
  float*  vattn = ws + WS_VATT;
  float*  out   = (float*)d_out;

  bn_partials_k<<<dim3(32, 128), 256, 0, stream>>>(a1, part);
  bn_finalize_k<<<1, 128, 0, stream>>>(part, w_val, b_val, g_val, be_val,
                                       w_gate, b_gate, g_gate, be_gate, coef);
  v_paths_k<<<4, 256, 0, stream>>>(v1, w_key, b_key, g_key, be_key,
                                   w_attn, b_attn, g_attn, be_attn, vkey, vattn);
  fuse_k<<<1024, 256, 0, stream>>>(a1, coef, vkey, vattn, out);
}